// SphereConv2d_18184891532105
// MI455X (gfx1250) — compile-verified
//
#include <hip/hip_runtime.h>

typedef __attribute__((ext_vector_type(2))) float v2f;
typedef __attribute__((ext_vector_type(8))) float v8f;
typedef __attribute__((ext_vector_type(4))) unsigned int v4u;
typedef __attribute__((ext_vector_type(4))) int v4i;
typedef __attribute__((ext_vector_type(8))) int v8i;

constexpr int kN    = 2;
constexpr int kInC  = 64;
constexpr int kOutC = 128;
constexpr int kH    = 256;
constexpr int kW    = 512;
constexpr int kKS   = 3;
constexpr int kHW   = kH * kW;
constexpr int kGrp  = kOutC / 16;                  // 8 out-channel groups
constexpr int kWElems = kOutC * kInC * kKS * kKS;  // 73728 weight floats (288 KB)
constexpr int kWPad   = kWElems + kWElems / 64;    // +1 dword pad per 64 -> 74880

#if defined(__has_builtin)
#if __has_builtin(__builtin_amdgcn_tensor_load_to_lds) && \
    __has_builtin(__builtin_amdgcn_s_wait_tensorcnt)
#define USE_TDM 1
#endif
#endif

// LDS index with TDM padding applied (1 dword inserted every 64 dwords).
// Lane stride for B reads becomes 585 = 9 (mod 64) -> 16 lanes, 16 banks.
__device__ __forceinline__ int pad_idx(int i) { return i + (i >> 6); }

// 512 threads = 16 wave32. One wave = one 16-pixel M-tile x all 128 out-ch.
// D(16x16) = A(16 pix x 4 ch) * B(4 ch x 16 outch) + C over K = 9*64 = 576
// via V_WMMA_F32_16X16X4_F32. Weights staged once per block by the TDM.
__global__ __launch_bounds__(512)
void sphere_conv_wmma_tdm(const float* __restrict__ x,
                          const float* __restrict__ wgt,
                          const float* __restrict__ bias,
                          const float* __restrict__ grid,
                          float* __restrict__ out)
{
    __shared__ float wlds[kWPad];                  // ~292.5 KB of the 320 KB WGP LDS

    const int lane = threadIdx.x & 31;
    const int wave = threadIdx.x >> 5;
    const int tile = blockIdx.x * 16 + wave;
    const int m0   = tile * 16;

    const int row  = lane & 15;                    // A row / output pixel in tile
    const int hi   = lane >> 4;                    // K-slot half: {0,1} vs {2,3}
    const int koff = hi << 1;

    // Flat LDS pointer low 32 bits == workgroup-relative LDS byte offset
    // (ISA 10.2 aperture mapping).
    const unsigned lds_base = (unsigned)(unsigned long long)(const void*)wlds;

    // ---- stage all conv weights into LDS with one Tensor-DMA ----
#ifdef USE_TDM
    if (threadIdx.x == 0) {
        const unsigned long long ga = (unsigned long long)(const void*)wgt;
        v4u g0;
        g0[0] = 1u;                                             // count=1 valid D#
        g0[1] = lds_base;                                       // lds_addr (bytes)
        g0[2] = (unsigned)(ga & 0xFFFFFFFFu);                   // global_addr lo
        g0[3] = (unsigned)(((ga >> 32) & 0x1FFFFFFu) | (2u << 30)); // hi | type=2
        v8i g1;
        g1[0] = (int)((2u << 16) |                              // data_size = 4B
                      (1u << 20) |                              // pad_enable
                      (5u << 22));                              // pad every 64 dw
                                                                // pad_amount=0 -> 1 dw
        g1[1] = (int)(4608u << 16);                             // tensor_dim0 = 4608
        g1[2] = (int)(16u << 16);                               // tensor_dim1 = 16
        g1[3] = (int)(4608u << 16);                             // tile_dim0 = 4608
        g1[4] = 16;                                             // tile_dim1 = 16
        g1[5] = 4608;                                           // dim0 stride (contig)
        g1[6] = 0;
        g1[7] = 0;
        v4i gz = {0, 0, 0, 0};
#if __clang_major__ >= 23
        v8i gz8 = {0, 0, 0, 0, 0, 0, 0, 0};
        __builtin_amdgcn_tensor_load_to_lds(g0, g1, gz, gz, gz8, 0);
#else
        __builtin_amdgcn_tensor_load_to_lds(g0, g1, gz, gz, 0);
#endif
        __builtin_amdgcn_s_wait_tensorcnt(0);
    }
#else
    for (int k = threadIdx.x; k < kWElems; k += (int)blockDim.x)
        wlds[pad_idx(k)] = wgt[k];
#endif
    __syncthreads();

    // The TDM wrote wlds behind the compiler's back: pointer operand + memory
    // clobber keeps alias analysis from folding the ds_loads below to undef.
    asm volatile("" : : "r"(wlds), "r"((unsigned long long)lds_base) : "memory");

    // ---- my pixel (lane L and L+16 share a pixel) ----
    const int p   = m0 + row;
    const int n   = p / kHW;
    const int rem = p - n * kHW;
    const int h   = rem / kW;
    const int w   = rem - h * kW;
    const float* xb = x + (long)n * kInC * kHW;

    // accumulators with bias folded in (bias depends only on N = g*16+row)
    v8f acc[kGrp];
#pragma unroll
    for (int g = 0; g < kGrp; ++g) {
        const float bv = bias[g * 16 + row];
#pragma unroll
        for (int r = 0; r < 8; ++r) acc[g][r] = bv;
    }

    // tap-major K ordering: bilinear setup hoisted out of the channel loop
    for (int t = 0; t < kKS * kKS; ++t) {
        const int ti = t / kKS;
        const int tj = t - ti * kKS;

        const long gidx = ((long)(h * kKS + ti) * (kW * kKS) + (w * kKS + tj)) * 2;
        const float gx = grid[gidx];
        const float gy = grid[gidx + 1];
        const float fix = ((gx + 1.0f) * kW - 1.0f) * 0.5f;
        const float fiy = ((gy + 1.0f) * kH - 1.0f) * 0.5f;
        const float x0 = floorf(fix);
        const float y0 = floorf(fiy);
        const float fx = fix - x0;
        const float fy = fiy - y0;

        int   off[4];
        float cw[4];
#pragma unroll
        for (int c = 0; c < 4; ++c) {
            const float xi = x0 + (float)(c & 1);
            const float yi = y0 + (float)(c >> 1);
            const float wc = ((c & 1) ? fx : 1.0f - fx) *
                             ((c >> 1) ? fy : 1.0f - fy);
            const bool valid = (xi >= 0.0f) && (xi <= (float)(kW - 1)) &&
                               (yi >= 0.0f) && (yi <= (float)(kH - 1));
            const int xc = (int)fminf(fmaxf(xi, 0.0f), (float)(kW - 1));
            const int yc = (int)fminf(fmaxf(yi, 0.0f), (float)(kH - 1));
            off[c] = yc * kW + xc;
            cw[c]  = valid ? wc : 0.0f;
        }

#pragma unroll 2
        for (int c0 = 0; c0 < kInC; c0 += 4) {
            const int cA = c0 + koff;              // this lane's two K slots
            const float* p0 = xb + (long)cA * kHW;
            const float* p1 = p0 + kHW;
            v2f a;
            a.x = cw[0] * p0[off[0]] + cw[1] * p0[off[1]] +
                  cw[2] * p0[off[2]] + cw[3] * p0[off[3]];
            a.y = cw[0] * p1[off[0]] + cw[1] * p1[off[1]] +
                  cw[2] * p1[off[2]] + cw[3] * p1[off[3]];

#pragma unroll
            for (int g = 0; g < kGrp; ++g) {
                const int o    = g * 16 + row;     // B column (out-channel)
                const int idx0 = (o * kInC + cA) * (kKS * kKS) + t;
                v2f b;
                b.x = wlds[pad_idx(idx0)];                 // channel cA
                b.y = wlds[pad_idx(idx0 + kKS * kKS)];     // channel cA+1
                acc[g] = __builtin_amdgcn_wmma_f32_16x16x4_f32(
                    /*neg_a=*/false, a, /*neg_b=*/false, b,
                    /*c_mod=*/(short)0, acc[g],
                    /*reuse_a=*/false, /*reuse_b=*/false);
            }
        }
    }

    // epilogue: C/D layout -> element (M = r + 8*hi, N = g*16 + row)
#pragma unroll
    for (int r = 0; r < 8; ++r) {
        const int M  = r + (hi << 3);
        const int pp = m0 + M;
        const int nn = pp / kHW;
        const int rr = pp - nn * kHW;
        const int hh = rr / kW;
        const int ww = rr - hh * kW;
        float* ob = out + (((long)nn * kOutC) * kH + hh) * kW + ww;
#pragma unroll
        for (int g = 0; g < kGrp; ++g) {
            const int o = g * 16 + row;
            ob[(long)o * kHW] = acc[g][r];
        }
    }
}

extern "C" void kernel_launch(void* const* d_in, const int* in_sizes, int n_in,
                              void* d_out, int out_size, void* d_ws, size_t ws_size,
                              hipStream_t stream)
{
    const float* x      = (const float*)d_in[0];
    const float* weight = (const float*)d_in[1];
    const float* bias   = (const float*)d_in[2];
    const float* grid   = (const float*)d_in[3];
    float*       out    = (float*)d_out;

    constexpr int totalM = kN * kH * kW;   // 262144 output pixels
    constexpr int tiles  = totalM / 16;    // 16384 wave tiles
    dim3 blk(512);                         // 16 wave32 per block
    dim3 grd(tiles / 16);                  // 1024 blocks
    hipLaunchKernelGGL(sphere_conv_wmma_tdm, grd, blk, 0, stream,
                       x, weight, bias, grid, out);
}